// AttentionConv_58548994179159
// MI455X (gfx1250) — compile-verified
//
#include <hip/hip_runtime.h>
#include <hip/hip_bf16.h>
#include <math.h>

// Problem constants (from reference): B=4, C=Cout=64, H=W=64, K=7, PAD=3.
#define BATCH 4
#define CH    64
#define HH    64
#define WW    64
#define HWSZ  (HH * WW)        // 4096
#define NTOT  (BATCH * HWSZ)   // 16384
#define KS    7
#define PADR  3

typedef __attribute__((ext_vector_type(2))) float v2f;
typedef __attribute__((ext_vector_type(8))) float v8f;

// ---------------------------------------------------------------------------
// Stage 1: 1x1-conv projections as fp32 WMMA GEMMs.
//   q = Wq * x, k = Wk * x, v = Wv * y
// One wave computes one 16(M) x 16(N) tile, sweeping K=64 as 16 steps of
// V_WMMA_F32_16X16X4_F32 (exact fp32 semantics — this problem is bandwidth
// bound, so there is zero reason to drop precision to f16).
//
// Fragment layouts (ISA 7.12.2, wave32):
//   A 16x4 f32 : lanes 0-15 -> M=lane,    VGPR0=K0, VGPR1=K1
//                lanes 16-31 -> M=lane-16, VGPR0=K2, VGPR1=K3
//   B 4x16 f32 : VGPR0: lanes0-15 row K0 / lanes16-31 row K2; VGPR1: K1/K3
//   C/D 16x16  : VGPR r: lanes0-15 (M=r, N=lane); lanes16-31 (M=r+8)
// ---------------------------------------------------------------------------
__global__ void __launch_bounds__(256)
proj_wmma(const float* __restrict__ x, const float* __restrict__ y,
          const float* __restrict__ Wq, const float* __restrict__ Wk,
          const float* __restrict__ Wv,
          float* __restrict__ qb, float* __restrict__ kb, float* __restrict__ vb) {
    const int lane = threadIdx.x & 31;
    const int wave = blockIdx.x * (blockDim.x >> 5) + (threadIdx.x >> 5);

    const int NT = NTOT / 16;              // 1024 N-tiles
    int t = wave;
    const int nt   = t % NT; t /= NT;
    const int mt   = t % 4;  t /= 4;       // 4 M-tiles (64 rows / 16)
    const int proj = t;                    // 0=q, 1=k, 2=v (wave-uniform)

    const float* Wmat = (proj == 0) ? Wq : ((proj == 1) ? Wk : Wv);
    const float* inp  = (proj == 2) ? y : x;
    float*       outp = (proj == 0) ? qb : ((proj == 1) ? kb : vb);

    const int m0  = mt * 16;
    const int n0  = nt * 16;
    const int bat = n0 >> 12;              // 4096 % 16 == 0 -> batch uniform in tile
    const int hw0 = n0 & (HWSZ - 1);
    const int nl  = lane & 15;
    const int hi  = lane >> 4;             // which K-half this lane holds

    const float* inb = inp + (size_t)bat * CH * HWSZ;
    const int arow = m0 + nl;

    v8f acc = {};
#pragma unroll
    for (int kk = 0; kk < 16; ++kk) {
        const int kbase = kk * 4 + hi * 2;
        v2f a;
        a.x = Wmat[arow * CH + kbase];
        a.y = Wmat[arow * CH + kbase + 1];
        v2f bf;
        bf.x = inb[(size_t)(kbase    ) * HWSZ + hw0 + nl];
        bf.y = inb[(size_t)(kbase + 1) * HWSZ + hw0 + nl];
        acc = __builtin_amdgcn_wmma_f32_16x16x4_f32(
            /*neg_a=*/false, a, /*neg_b=*/false, bf,
            /*c_mod=*/(short)0, acc, /*reuse_a=*/false, /*reuse_b=*/false);
    }

    float* ob = outp + (size_t)bat * CH * HWSZ + hw0 + nl;
#pragma unroll
    for (int r = 0; r < 8; ++r) {
        const int row = m0 + r + hi * 8;
        ob[(size_t)row * HWSZ] = acc[r];
    }
}

// ---------------------------------------------------------------------------
// Stage 2: per-channel 7x7 windowed softmax-attention, LDS-tiled.
// Block = 256 threads = 16x16 spatial tile of one (b, c) plane.
// k/v halo (22x22) staged in LDS (stride 23 vs 64 banks), online softmax.
// Padding semantics match reference: OOB -> k=0 (bias still added), v=0.
// ---------------------------------------------------------------------------
#define TILE 16
#define HALO (TILE + 2 * PADR)   // 22
#define LSTR (HALO + 1)          // 23

__global__ void __launch_bounds__(256)
attn_tiled(const float* __restrict__ qb, const float* __restrict__ kb,
           const float* __restrict__ vb, const float* __restrict__ rel_h,
           const float* __restrict__ rel_w, float* __restrict__ out) {
    __shared__ float ks[HALO * LSTR];
    __shared__ float vs[HALO * LSTR];

    int id = blockIdx.x;
    const int tw = id & 3;  id >>= 2;      // W/16 = 4 tiles
    const int th = id & 3;  id >>= 2;      // H/16 = 4 tiles
    const int c  = id & (CH - 1); id >>= 6;
    const int bat = id;                    // 0..3

    const int h0 = th * TILE;
    const int w0 = tw * TILE;
    const size_t plane = ((size_t)bat * CH + c) * HWSZ;
    const float* kp = kb + plane;
    const float* vp = vb + plane;

    // Stage k/v halo into LDS (zero outside the image == zero padding).
    for (int i = threadIdx.x; i < HALO * HALO; i += 256) {
        const int r  = i / HALO;
        const int cc = i - r * HALO;
        const int hh = h0 + r  - PADR;
        const int ww = w0 + cc - PADR;
        float kv = 0.0f, vv = 0.0f;
        if (hh >= 0 && hh < HH && ww >= 0 && ww < WW) {
            kv = kp[hh * WW + ww];
            vv = vp[hh * WW + ww];
        }
        ks[r * LSTR + cc] = kv;
        vs[r * LSTR + cc] = vv;
    }
    __syncthreads();

    const int tx = threadIdx.x & 15;
    const int ty = threadIdx.x >> 4;
    const int h = h0 + ty;
    const int w = w0 + tx;

    const float qv = qb[plane + h * WW + w];

    // bias[c, t] = (c < 32) ? rel_h[c][t/7] : rel_w[c-32][t%7]
    const bool byRow = (c < CH / 2);
    float bias7[KS];
    const float* relp = byRow ? (rel_h + c * KS) : (rel_w + (c - CH / 2) * KS);
#pragma unroll
    for (int i = 0; i < KS; ++i) bias7[i] = relp[i];

    // Online softmax over the 49 taps.
    float m = -INFINITY, s = 0.0f, o = 0.0f;
#pragma unroll
    for (int kh = 0; kh < KS; ++kh) {
#pragma unroll
        for (int kw = 0; kw < KS; ++kw) {
            const int li = (ty + kh) * LSTR + (tx + kw);
            const float kval = ks[li];
            const float vval = vs[li];
            const float bv = byRow ? bias7[kh] : bias7[kw];
            const float l = qv * (kval + bv);
            const float mn = fmaxf(m, l);
            const float e0 = __expf(m - mn);   // first iter: exp(-inf)=0
            const float e1 = __expf(l - mn);
            s = s * e0 + e1;
            o = o * e0 + e1 * vval;
            m = mn;
        }
    }
    out[plane + h * WW + w] = o / s;
}

// ---------------------------------------------------------------------------
extern "C" void kernel_launch(void* const* d_in, const int* in_sizes, int n_in,
                              void* d_out, int out_size, void* d_ws, size_t ws_size,
                              hipStream_t stream) {
    const float* x     = (const float*)d_in[0];
    const float* y     = (const float*)d_in[1];
    const float* Wq    = (const float*)d_in[2];
    const float* Wk    = (const float*)d_in[3];
    const float* Wv    = (const float*)d_in[4];
    const float* rel_h = (const float*)d_in[5];
    const float* rel_w = (const float*)d_in[6];
    float* out = (float*)d_out;

    // Workspace: q, k, v planes (each B*C*H*W fp32 = 4 MiB; 12 MiB total).
    float* qb = (float*)d_ws;
    float* kb = qb + (size_t)BATCH * CH * HWSZ;
    float* vb = kb + (size_t)BATCH * CH * HWSZ;

    // 3 projections * 4 M-tiles * 1024 N-tiles = 12288 waves; 8 waves/block.
    const int waves = 3 * 4 * (NTOT / 16);
    proj_wmma<<<waves / 8, 256, 0, stream>>>(x, y, Wq, Wk, Wv, qb, kb, vb);

    // B * C * (H/16) * (W/16) = 4096 blocks.
    attn_tiled<<<BATCH * CH * (HH / TILE) * (WW / TILE), 256, 0, stream>>>(
        qb, kb, vb, rel_h, rel_w, out);
}